// HMCLayer_89275190214713
// MI455X (gfx1250) — compile-verified
//
#include <hip/hip_runtime.h>
#include <hip/hip_bf16.h>

typedef __attribute__((ext_vector_type(2))) float v2f;
typedef __attribute__((ext_vector_type(8))) float v8f;

#define NEG_SLOPE 0.2f

// ---------------------------------------------------------------------------
// Zero-fill helper (graph-capture safe; no hipMemsetAsync dependence)
// ---------------------------------------------------------------------------
__global__ __launch_bounds__(256) void zero_kernel(float* __restrict__ p, size_t n) {
    size_t i = (size_t)blockIdx.x * blockDim.x + threadIdx.x;
    size_t stride = (size_t)gridDim.x * blockDim.x;
    for (; i < n; i += stride) p[i] = 0.0f;
}

// ---------------------------------------------------------------------------
// FP32 WMMA GEMM: Y[M x 64] = X[M x 128] @ W[128 x 64]
// One wave computes a 16-row x 64-col tile using V_WMMA_F32_16X16X4_F32.
// A layout  (16x4):  lane {g=lane>>4, i=lane&15}, vgpr r -> A[M=i][K=2g+r]
// B layout  (4x16):  vgpr r -> B[K=2g+r][N=i]
// C/D layout(16x16): vgpr r -> D[M=r+8g][N=i]
// ---------------------------------------------------------------------------
__global__ __launch_bounds__(256) void gemm_wmma_f32(const float* __restrict__ X,
                                                     const float* __restrict__ W,
                                                     float* __restrict__ Y,
                                                     int ntiles) {
    int wave = (int)((blockIdx.x * blockDim.x + threadIdx.x) >> 5);
    int lane = threadIdx.x & 31;
    if (wave >= ntiles) return;  // wave-uniform: EXEC stays all-1s for live waves

    int i = lane & 15;   // M (for A) / N (for B,C)
    int g = lane >> 4;   // K lane-group selector

    const float* xrow = X + ((size_t)wave * 16 + (size_t)i) * 128 + 2 * g;
    const float* wp   = W + (size_t)(2 * g) * 64 + i;

    v8f acc0 = {}, acc1 = {}, acc2 = {}, acc3 = {};

#pragma unroll 4
    for (int k0 = 0; k0 < 128; k0 += 4) {
        v2f a;
        a.x = xrow[k0];
        a.y = xrow[k0 + 1];

        const float* wk = wp + (size_t)k0 * 64;
        v2f b0, b1, b2, b3;
        b0.x = wk[0];  b0.y = wk[64];
        b1.x = wk[16]; b1.y = wk[80];
        b2.x = wk[32]; b2.y = wk[96];
        b3.x = wk[48]; b3.y = wk[112];

        acc0 = __builtin_amdgcn_wmma_f32_16x16x4_f32(false, a, false, b0, (short)0, acc0, false, false);
        acc1 = __builtin_amdgcn_wmma_f32_16x16x4_f32(false, a, false, b1, (short)0, acc1, false, false);
        acc2 = __builtin_amdgcn_wmma_f32_16x16x4_f32(false, a, false, b2, (short)0, acc2, false, false);
        acc3 = __builtin_amdgcn_wmma_f32_16x16x4_f32(false, a, false, b3, (short)0, acc3, false, false);
    }

    float* yb = Y + (size_t)wave * 16 * 64;
#pragma unroll
    for (int r = 0; r < 8; ++r) {
        int m = r + 8 * g;
        yb[m * 64 + 0  + i] = acc0[r];
        yb[m * 64 + 16 + i] = acc1[r];
        yb[m * 64 + 32 + i] = acc2[r];
        yb[m * 64 + 48 + i] = acc3[r];
    }
}

// ---------------------------------------------------------------------------
// score[m] = dot(msg[m, 0:64], att[off:off+64]); one wave32 per row
// ---------------------------------------------------------------------------
__global__ __launch_bounds__(256) void score_kernel(const float* __restrict__ msg,
                                                    const float* __restrict__ att,
                                                    int att_off,
                                                    float* __restrict__ score,
                                                    int M) {
    int wave = (int)((blockIdx.x * blockDim.x + threadIdx.x) >> 5);
    int lane = threadIdx.x & 31;
    if (wave >= M) return;
    const float* rowp = msg + (size_t)wave * 64;
    float v = rowp[lane] * att[att_off + lane] + rowp[lane + 32] * att[att_off + lane + 32];
#pragma unroll
    for (int s = 16; s > 0; s >>= 1) v += __shfl_xor(v, s, 32);
    if (lane == 0) score[wave] = v;
}

// ---------------------------------------------------------------------------
// Per-edge: edge = leaky_relu(s_score[col] + t_score[row]); segment sums
// ---------------------------------------------------------------------------
__global__ __launch_bounds__(256) void edge_kernel(const int* __restrict__ row_idx,
                                                   const int* __restrict__ col_idx,
                                                   const float* __restrict__ s_score,
                                                   const float* __restrict__ t_score,
                                                   float* __restrict__ edge,
                                                   float* __restrict__ e_row_sum,
                                                   float* __restrict__ f_row_sum,
                                                   int nnz) {
    int e = blockIdx.x * blockDim.x + threadIdx.x;
    if (e >= nnz) return;
    int r = row_idx[e];
    int c = col_idx[e];
    float x = s_score[c] + t_score[r];
    float ed = (x >= 0.0f) ? x : NEG_SLOPE * x;
    edge[e] = ed;
    atomicAdd(&e_row_sum[r], ed);
    atomicAdd(&f_row_sum[c], ed);
}

// ---------------------------------------------------------------------------
// Per-edge scatter of 64-wide message rows. One wave per edge; each lane
// covers 2 of the 64 columns in each direction (4 float atomics / lane).
// ---------------------------------------------------------------------------
__global__ __launch_bounds__(256) void scatter_kernel(const int* __restrict__ row_idx,
                                                      const int* __restrict__ col_idx,
                                                      const float* __restrict__ nbhd,
                                                      const float* __restrict__ edge,
                                                      const float* __restrict__ e_row_sum,
                                                      const float* __restrict__ f_row_sum,
                                                      const float* __restrict__ s_msg,
                                                      const float* __restrict__ t_msg,
                                                      float* __restrict__ msg_src,   // n_s x 64
                                                      float* __restrict__ msg_tgt,   // n_t x 64
                                                      int nnz) {
    int e    = (int)((blockIdx.x * blockDim.x + threadIdx.x) >> 5);
    int lane = threadIdx.x & 31;
    if (e >= nnz) return;

    int r = row_idx[e];
    int c = col_idx[e];
    float ed = edge[e];
    float nb = nbhd[e];

    float es = e_row_sum[r]; es = (es == 0.0f) ? 1.0f : es;
    float fs = f_row_sum[c]; fs = (fs == 0.0f) ? 1.0f : fs;
    float ev = (ed / es) * nb;  // weight for s_msg[c] -> msg_tgt[r]
    float fv = (ed / fs) * nb;  // weight for t_msg[r] -> msg_src[c]

    const float* smp = s_msg + (size_t)c * 64;
    const float* tmp = t_msg + (size_t)r * 64;
    float s0 = smp[lane], s1 = smp[lane + 32];
    float t0 = tmp[lane], t1 = tmp[lane + 32];

    float* dt = msg_tgt + (size_t)r * 64;
    float* ds = msg_src + (size_t)c * 64;
    atomicAdd(&dt[lane],      ev * s0);
    atomicAdd(&dt[lane + 32], ev * s1);
    atomicAdd(&ds[lane],      fv * t0);
    atomicAdd(&ds[lane + 32], fv * t1);
}

// ---------------------------------------------------------------------------
// Host launcher
// ---------------------------------------------------------------------------
extern "C" void kernel_launch(void* const* d_in, const int* in_sizes, int n_in,
                              void* d_out, int out_size, void* d_ws, size_t ws_size,
                              hipStream_t stream) {
    const float* x_source = (const float*)d_in[0];
    const float* x_target = (const float*)d_in[1];
    const float* nbhd     = (const float*)d_in[2];
    const float* w_s      = (const float*)d_in[3];
    const float* w_t      = (const float*)d_in[4];
    const float* att      = (const float*)d_in[5];  // 128 floats: a_s | a_t
    const int*   row_idx  = (const int*)d_in[6];
    const int*   col_idx  = (const int*)d_in[7];

    const int n_s = in_sizes[0] / 128;   // 100000
    const int n_t = in_sizes[1] / 128;   // 20000
    const int nnz = in_sizes[2];         // 2000000

    // Workspace layout (floats)
    float* ws        = (float*)d_ws;
    float* s_msg     = ws;                                   // n_s*64
    float* t_msg     = s_msg + (size_t)n_s * 64;             // n_t*64
    float* s_score   = t_msg + (size_t)n_t * 64;             // n_s
    float* t_score   = s_score + n_s;                        // n_t
    float* edge      = t_score + n_t;                        // nnz
    float* e_row_sum = edge + nnz;                           // n_t
    float* f_row_sum = e_row_sum + n_t;                      // n_s

    float* msg_src = (float*)d_out;                          // n_s*64 (first output)
    float* msg_tgt = msg_src + (size_t)n_s * 64;             // n_t*64 (second output)

    // 1) Zero the segment sums and the output accumulators
    {
        size_t nsum = (size_t)n_t + (size_t)n_s;
        int blk = (int)((nsum + 255) / 256);
        if (blk > 4096) blk = 4096;
        zero_kernel<<<blk, 256, 0, stream>>>(e_row_sum, nsum);

        size_t nout = (size_t)out_size;
        int blk2 = (int)((nout + 255) / 256);
        if (blk2 > 16384) blk2 = 16384;
        zero_kernel<<<blk2, 256, 0, stream>>>(msg_src, nout);
    }

    // 2) Projections via fp32 WMMA (M divisible by 16 for these sizes)
    {
        int tiles_s = n_s / 16;                       // 6250
        int tiles_t = n_t / 16;                       // 1250
        gemm_wmma_f32<<<(tiles_s + 7) / 8, 256, 0, stream>>>(x_source, w_s, s_msg, tiles_s);
        gemm_wmma_f32<<<(tiles_t + 7) / 8, 256, 0, stream>>>(x_target, w_t, t_msg, tiles_t);
    }

    // 3) Attention scores (wave per row)
    {
        int waves_s = n_s, waves_t = n_t;
        score_kernel<<<(waves_s * 32 + 255) / 256, 256, 0, stream>>>(s_msg, att, 0,  s_score, n_s);
        score_kernel<<<(waves_t * 32 + 255) / 256, 256, 0, stream>>>(t_msg, att, 64, t_score, n_t);
    }

    // 4) Edge values + segment sums
    edge_kernel<<<(nnz + 255) / 256, 256, 0, stream>>>(row_idx, col_idx, s_score, t_score,
                                                       edge, e_row_sum, f_row_sum, nnz);

    // 5) Normalized message scatter (wave per edge)
    {
        long long threads = (long long)nnz * 32;
        int blocks = (int)((threads + 255) / 256);
        scatter_kernel<<<blocks, 256, 0, stream>>>(row_idx, col_idx, nbhd, edge,
                                                   e_row_sum, f_row_sum, s_msg, t_msg,
                                                   msg_src, msg_tgt, nnz);
    }
}